// PropConv_12266426598060
// MI455X (gfx1250) — compile-verified
//
#include <hip/hip_runtime.h>

#define C_TOTAL 256
#define C_HALF  128
#define EDGES_PER_WAVE 4

__device__ __forceinline__ int   rfl_i(int x)   { return __builtin_amdgcn_readfirstlane(x); }
__device__ __forceinline__ float rfl_f(float x) {
    return __int_as_float(__builtin_amdgcn_readfirstlane(__float_as_int(x)));
}

// ---------------------------------------------------------------------------
// Zero-fill with 128-bit stores (harness poisons d_out / d_ws with 0xAA).
// ---------------------------------------------------------------------------
__global__ __launch_bounds__(256) void zero_f4_kernel(float4* __restrict__ p, int n4) {
    int i = blockIdx.x * blockDim.x + threadIdx.x;
    if (i < n4) {
        p[i] = make_float4(0.f, 0.f, 0.f, 0.f);
    }
}

// ---------------------------------------------------------------------------
// Scatter: one wave32 per EDGES_PER_WAVE (4) edges. Lane l owns channels [4l, 4l+4).
//  fwd: out[row, 0:128]   += w * x[col, 0:128]
//  bwd: out[col, 128:256] += w * x[row, 128:256]
// Fast path batches: (1) 12 scalar metadata loads, (2) 8 saddr-form b128
// gathers in flight, (3) 32 no-return f32 atomics (+8 degree atomics on lane 0) — maximizing MLP per wave.
// ---------------------------------------------------------------------------
__global__ __launch_bounds__(256) void propconv_scatter_kernel(
    const float* __restrict__ x,
    const float* __restrict__ ew,
    const int*   __restrict__ row,
    const int*   __restrict__ col,
    float*       __restrict__ out,
    float*       __restrict__ deg_row,
    float*       __restrict__ deg_col,
    int n_edges)
{
    const int lane = threadIdx.x & 31;
    const int wave = blockIdx.x * (blockDim.x >> 5) + (threadIdx.x >> 5);
    const int base = rfl_i(wave) * EDGES_PER_WAVE;   // wave-uniform -> SGPR
    const int ch   = lane * 4;

    // Streaming prefetch of edge metadata ~2KB ahead (global_prefetch_b8).
    if (lane == 0 && base + 512 < n_edges) {
        __builtin_prefetch(row + base + 512, 0, 0);
        __builtin_prefetch(col + base + 512, 0, 0);
        __builtin_prefetch(ew  + base + 512, 0, 0);
    }

    if (base + EDGES_PER_WAVE <= n_edges) {
        // ---- fast path: full tile, batched phases ----
        int   r[EDGES_PER_WAVE], c[EDGES_PER_WAVE];
        float wt[EDGES_PER_WAVE];
#pragma unroll
        for (int k = 0; k < EDGES_PER_WAVE; ++k) {
            r[k]  = rfl_i(row[base + k]);
            c[k]  = rfl_i(col[base + k]);
            wt[k] = rfl_f(ew[base + k]);
        }

        float4 xf[EDGES_PER_WAVE], xb[EDGES_PER_WAVE];
#pragma unroll
        for (int k = 0; k < EDGES_PER_WAVE; ++k) {
            xf[k] = *(const float4*)(x + c[k] * C_TOTAL + ch);
            xb[k] = *(const float4*)(x + r[k] * C_TOTAL + C_HALF + ch);
        }

#pragma unroll
        for (int k = 0; k < EDGES_PER_WAVE; ++k) {
            float* of = out + r[k] * C_TOTAL + ch;           // fwd -> row
            float* ob = out + c[k] * C_TOTAL + C_HALF + ch;  // bwd -> col
            atomicAdd(of + 0, wt[k] * xf[k].x);
            atomicAdd(of + 1, wt[k] * xf[k].y);
            atomicAdd(of + 2, wt[k] * xf[k].z);
            atomicAdd(of + 3, wt[k] * xf[k].w);
            atomicAdd(ob + 0, wt[k] * xb[k].x);
            atomicAdd(ob + 1, wt[k] * xb[k].y);
            atomicAdd(ob + 2, wt[k] * xb[k].z);
            atomicAdd(ob + 3, wt[k] * xb[k].w);
        }

        if (lane == 0) {
#pragma unroll
            for (int k = 0; k < EDGES_PER_WAVE; ++k) {
                atomicAdd(deg_row + r[k], 1.0f);
                atomicAdd(deg_col + c[k], 1.0f);
            }
        }
    } else {
        // ---- tail: per-edge guarded ----
        for (int k = 0; k < EDGES_PER_WAVE; ++k) {
            const int e = base + k;
            if (e >= n_edges) return;
            const int   r  = rfl_i(row[e]);
            const int   c  = rfl_i(col[e]);
            const float w  = rfl_f(ew[e]);

            const float4 xf = *(const float4*)(x + c * C_TOTAL + ch);
            const float4 xb = *(const float4*)(x + r * C_TOTAL + C_HALF + ch);

            float* of = out + r * C_TOTAL + ch;
            float* ob = out + c * C_TOTAL + C_HALF + ch;
            atomicAdd(of + 0, w * xf.x);
            atomicAdd(of + 1, w * xf.y);
            atomicAdd(of + 2, w * xf.z);
            atomicAdd(of + 3, w * xf.w);
            atomicAdd(ob + 0, w * xb.x);
            atomicAdd(ob + 1, w * xb.y);
            atomicAdd(ob + 2, w * xb.z);
            atomicAdd(ob + 3, w * xb.w);

            if (lane == 0) {
                atomicAdd(deg_row + r, 1.0f);
                atomicAdd(deg_col + c, 1.0f);
            }
        }
    }
}

// ---------------------------------------------------------------------------
// Normalize: out[n, 0:128]   /= max(deg_row[n], 1)
//            out[n, 128:256] /= max(deg_col[n], 1)
// One float4 per thread; each wave covers exactly half a node's channels,
// so n and the deg selection are wave-uniform -> scalar broadcast load.
// ---------------------------------------------------------------------------
__global__ __launch_bounds__(256) void propconv_norm_kernel(
    float*       __restrict__ out,
    const float* __restrict__ deg_row,
    const float* __restrict__ deg_col,
    int n_nodes)
{
    const int i = blockIdx.x * blockDim.x + threadIdx.x;
    const int n = rfl_i(i >> 6);            // node (uniform per wave)
    if (n >= n_nodes) return;
    const int g = i & 63;                   // float4 group within node (0..63)

    float d = (g < 32) ? deg_row[n] : deg_col[n];   // uniform select + broadcast
    d = fmaxf(d, 1.0f);

    float4* p = (float4*)(out + (size_t)n * C_TOTAL) + g;
    float4 v = *p;
    v.x /= d;
    v.y /= d;
    v.z /= d;
    v.w /= d;
    *p = v;
}

// ---------------------------------------------------------------------------
// Launch
//   d_in[0] = x           (n_nodes * 256 f32)
//   d_in[1] = edge_weight (n_edges f32)
//   d_in[2] = edge_index  (2 * n_edges int32: row then col)
//   d_out   = (n_nodes * 256 f32)
//   d_ws    = deg_row (n_nodes f32) | deg_col (n_nodes f32)
// ---------------------------------------------------------------------------
extern "C" void kernel_launch(void* const* d_in, const int* in_sizes, int n_in,
                              void* d_out, int out_size, void* d_ws, size_t ws_size,
                              hipStream_t stream) {
    const float* x  = (const float*)d_in[0];
    const float* ew = (const float*)d_in[1];
    const int*   ei = (const int*)d_in[2];

    const int n_edges = in_sizes[1];
    const int n_nodes = in_sizes[0] / C_TOTAL;

    const int* row = ei;
    const int* col = ei + n_edges;

    float* out     = (float*)d_out;
    float* deg_row = (float*)d_ws;
    float* deg_col = deg_row + n_nodes;

    // 1) zero output + degree scratch
    const int out4 = out_size / 4;
    zero_f4_kernel<<<(out4 + 255) / 256, 256, 0, stream>>>((float4*)out, out4);
    const int deg4 = (2 * n_nodes + 3) / 4;
    zero_f4_kernel<<<(deg4 + 255) / 256, 256, 0, stream>>>((float4*)d_ws, deg4);

    // 2) edge scatter: EDGES_PER_WAVE edges per wave
    const int waves  = (n_edges + EDGES_PER_WAVE - 1) / EDGES_PER_WAVE;
    const int blocks = (waves + 7) / 8;       // 8 wave32 per 256-thread block
    propconv_scatter_kernel<<<blocks, 256, 0, stream>>>(
        x, ew, row, col, out, deg_row, deg_col, n_edges);

    // 3) per-node normalization
    const int total = n_nodes * (C_TOTAL / 4);
    propconv_norm_kernel<<<(total + 255) / 256, 256, 0, stream>>>(
        out, deg_row, deg_col, n_nodes);
}